// Sineconv_6262062318192
// MI455X (gfx1250) — compile-verified
//
#include <hip/hip_runtime.h>
#include <math.h>

// ---------------------------------------------------------------------------
// Sineconv on MI455X (gfx1250).
//   out[b,l,f] = sum_w x[b, l + 2f + ((f+w)>>5)] * amp_f*sin(2pi*(tn_f*(l+w)+pn_f))
//              + x[b, l+32] * rk[f]
// Collapsed to: per output 1 double-fma phase (in revolutions), frac, hw
// V_SIN/V_COS (revolution-domain), 3-term dot against precomputed segment
// sums, and a WMMA-fused rank-1 residual.
// CDNA5 paths: GLOBAL_LOAD_ASYNC_TO_LDS_B32 (+ s_wait_asynccnt),
// V_WMMA_F32_16X16X4_F32 (x2, unrolled), v_sin_f32/v_cos_f32.
// ---------------------------------------------------------------------------

#define B_DIM   4
#define S_DIM   16384
#define F_DIM   32
#define W_DIM   65
#define L_OUT   16320          // S - W + 1
#define TILE_L  64             // per block (4 waves x 16 l)

typedef __attribute__((ext_vector_type(2))) float v2f;
typedef __attribute__((ext_vector_type(8))) float v8f;

#if defined(__HIP_DEVICE_COMPILE__) && __has_builtin(__builtin_amdgcn_wmma_f32_16x16x4_f32)
#define HAVE_WMMA4 1
#else
#define HAVE_WMMA4 0
#endif

#if defined(__HIP_DEVICE_COMPILE__) && __has_builtin(__builtin_amdgcn_global_load_async_to_lds_b32)
#define HAVE_ASYNC 1
#else
#define HAVE_ASYNC 0
#endif

#if HAVE_ASYNC
typedef __attribute__((address_space(1))) int gint_t;   // global
typedef __attribute__((address_space(3))) int lint_t;   // LDS
#endif

__device__ __forceinline__ void wait_async_zero() {
#if defined(__HIP_DEVICE_COMPILE__)
#if __has_builtin(__builtin_amdgcn_s_wait_asynccnt)
  __builtin_amdgcn_s_wait_asynccnt(0);
#else
  asm volatile("s_wait_asynccnt 0" ::: "memory");
#endif
#endif
}

// sin/cos of 2*pi*r for r in [0,1): hardware V_SIN_F32 / V_COS_F32 take
// revolutions (OCML native_sin = __builtin_amdgcn_sinf(x/(2*pi))).
__device__ __forceinline__ float sin_rev(float r) {
#if defined(__HIP_DEVICE_COMPILE__) && __has_builtin(__builtin_amdgcn_sinf)
  return __builtin_amdgcn_sinf(r);
#else
  return sinf(6.28318530717958647692f * r);
#endif
}
__device__ __forceinline__ float cos_rev(float r) {
#if defined(__HIP_DEVICE_COMPILE__) && __has_builtin(__builtin_amdgcn_cosf)
  return __builtin_amdgcn_cosf(r);
#else
  return cosf(6.28318530717958647692f * r);
#endif
}

// ---------------------------------------------------------------------------
// Kernel 1: per-feature constants (double precision, one wave, runs once).
//   tnd[f] = frequency*1.1875 (revolutions/sample), pnd[f] = phases[f]
//   fc[f*8 + {0..7}] = amp*Cc0..2, amp*Cs0..2, rk, pad
//   Ccj = sum_{w in seg j(f)} cos(2pi*tn*w), segments j = (f+w)>>5.
// ---------------------------------------------------------------------------
__global__ void sineconv_setup(const float* __restrict__ phases,
                               const float* __restrict__ amplitude,
                               const float* __restrict__ frequency,
                               const float* __restrict__ res_kernel,
                               double* __restrict__ tnd,
                               double* __restrict__ pnd,
                               float* __restrict__ fc) {
  int f = threadIdx.x;
  if (f >= F_DIM) return;
  const double TWO_PI = 6.283185307179586476925286766559;
  double tn  = (double)frequency[f] * 1.1875;   // 19000/16000 exact
  double amp = (double)amplitude[f];
  double Cc[3] = {0.0, 0.0, 0.0};
  double Cs[3] = {0.0, 0.0, 0.0};
  for (int w = 0; w < W_DIM; ++w) {
    int j = (f + w) >> 5;                        // 0,1,2
    double a = tn * (double)w;
    a -= floor(a);                               // frac -> [0,1)
    Cc[j] += cos(TWO_PI * a);
    Cs[j] += sin(TWO_PI * a);
  }
  tnd[f] = tn;
  pnd[f] = (double)phases[f];
  float* o = fc + f * 8;
  o[0] = (float)(amp * Cc[0]); o[1] = (float)(amp * Cc[1]); o[2] = (float)(amp * Cc[2]);
  o[3] = (float)(amp * Cs[0]); o[4] = (float)(amp * Cs[1]); o[5] = (float)(amp * Cs[2]);
  o[6] = res_kernel[f];
  o[7] = 0.0f;
}

// ---------------------------------------------------------------------------
// Kernel 2: fused main kernel. Block = 128 threads = 4 waves; each wave owns a
// 16(l) x 32(f) output strip -> two WMMA 16x16 tiles. Grid = (255, 4).
// Fully branch-free -> EXEC all-ones at every WMMA (ISA requirement).
// ---------------------------------------------------------------------------
__global__ __launch_bounds__(128) void sineconv_main(const float* __restrict__ x,
                                                     const double* __restrict__ tnd,
                                                     const double* __restrict__ pnd,
                                                     const float* __restrict__ fc,
                                                     float* __restrict__ out) {
  __shared__ float xs[128];   // x[b, L0 .. L0+127]; max index needed = L0+63+62+2

  const int b   = blockIdx.y;
  const int L0  = blockIdx.x * TILE_L;
  const int tid = threadIdx.x;

  const float* gx = x + (size_t)b * S_DIM + L0;

#if HAVE_ASYNC
  // CDNA5 async copy: one global_load_async_to_lds_b32 per wave (per-lane b32).
  __builtin_amdgcn_global_load_async_to_lds_b32(
      (gint_t*)(gx + tid),
      (lint_t*)(&xs[tid]),
      0, 0);
  wait_async_zero();
#else
  xs[tid] = gx[tid];
#endif
  __syncthreads();

  const int wv   = tid >> 5;        // wave 0..3
  const int lane = tid & 31;
  const int n    = lane & 15;       // f within 16-group == WMMA N
  const int hi   = lane >> 4;       // selects M+8 half of C/D layout
  const int lloc = wv * 16;         // LDS-relative l base of this wave
  const int l0w  = L0 + lloc;       // global l base of this wave

  // Arithmetic lane mask (no EXEC manipulation): 1.0 for lanes 0-15, else 0.
  const float lomask = (hi == 0) ? 1.0f : 0.0f;

#if HAVE_WMMA4
  // A operand is the same for both f-halves: K=0 column = x[l0w + M + 32]
  // (lanes 0-15, VGPR0); all other A entries zeroed via lomask.
  v2f a;
  a.x = xs[lloc + lane + 32] * lomask;   // index <= 48+31+32 = 111, always valid
  a.y = 0.0f;
#endif

  const size_t outb = ((size_t)b * L_OUT + (size_t)(l0w + 8 * hi)) * F_DIM;

#pragma unroll
  for (int half = 0; half < 2; ++half) {
    const int f = half * 16 + n;
    const double tn = tnd[f];
    const double pn = pnd[f];
    const float* cf = fc + f * 8;
    const float cc0 = cf[0], cc1 = cf[1], cc2 = cf[2];
    const float cs0 = cf[3], cs1 = cf[4], cs2 = cf[5];
    const float rk  = cf[6];

    const int xoff  = lloc + 2 * f + 8 * hi;   // + v (+j) ; max 127
    const int lbase = l0w + 8 * hi;

    v8f acc = {};
#pragma unroll
    for (int v = 0; v < 8; ++v) {
      const float x0 = xs[xoff + v];
      const float x1 = xs[xoff + v + 1];
      const float x2 = xs[xoff + v + 2];
      // phase in revolutions: exact in double, then frac -> hw sin/cos domain
      double rev = fma(tn, (double)(lbase + v), pn);
      rev -= floor(rev);
      const float r = (float)rev;
      const float s = sin_rev(r);
      const float c = cos_rev(r);
      const float rc = fmaf(x2, cc2, fmaf(x1, cc1, x0 * cc0));
      const float rs = fmaf(x2, cs2, fmaf(x1, cs1, x0 * cs0));
      acc[v] = fmaf(s, rc, c * rs);
    }

#if HAVE_WMMA4
    // Residual rank-1 update via WMMA: D = A(16x4) * B(4x16) + C.
    // B: K=0 row = rk[f0 + N] (lanes 0-15, VGPR0); upper lanes zeroed by mask.
    v2f bm;
    bm.x = rk * lomask;                  // unconditional load, masked value
    bm.y = 0.0f;
    v8f d = __builtin_amdgcn_wmma_f32_16x16x4_f32(
        /*neg_a=*/false, a, /*neg_b=*/false, bm,
        /*c_mod=*/(short)0, acc, /*reuse_a=*/false, /*reuse_b=*/false);
#else
    v8f d;
#pragma unroll
    for (int v = 0; v < 8; ++v)
      d[v] = fmaf(xs[lloc + 8 * hi + v + 32], rk, acc[v]);
#endif

    float* po = out + outb + f;
#pragma unroll
    for (int v = 0; v < 8; ++v)
      po[(size_t)v * F_DIM] = d[v];
  }
}

// ---------------------------------------------------------------------------
// Host launcher. Input order: x, sine_range(unused; == arange/16000 analytic),
// phases, amplitude, frequency, res_kernel. Output: (4, 16320, 32) f32.
// Workspace layout: 32 doubles tn | 32 doubles pn | 32*8 floats  (1536 B).
// ---------------------------------------------------------------------------
extern "C" void kernel_launch(void* const* d_in, const int* in_sizes, int n_in,
                              void* d_out, int out_size, void* d_ws, size_t ws_size,
                              hipStream_t stream) {
  (void)in_sizes; (void)n_in; (void)out_size; (void)ws_size;
  const float* x          = (const float*)d_in[0];
  const float* phases     = (const float*)d_in[2];
  const float* amplitude  = (const float*)d_in[3];
  const float* frequency  = (const float*)d_in[4];
  const float* res_kernel = (const float*)d_in[5];

  double* tnd = (double*)d_ws;
  double* pnd = tnd + F_DIM;
  float*  fc  = (float*)(pnd + F_DIM);
  float*  out = (float*)d_out;

  sineconv_setup<<<1, 32, 0, stream>>>(phases, amplitude, frequency, res_kernel,
                                       tnd, pnd, fc);

  dim3 grid(L_OUT / TILE_L, B_DIM);     // (255, 4)
  sineconv_main<<<grid, 128, 0, stream>>>(x, tnd, pnd, fc, out);
}